// AddMaskHead_34325378629883
// MI455X (gfx1250) — compile-verified
//
#include <hip/hip_runtime.h>
#include <hip/hip_bf16.h>

typedef __attribute__((ext_vector_type(2))) float v2f;
typedef __attribute__((ext_vector_type(8))) float v8f;

#define PR 14
#define CIN 512
#define CMID 256
#define KTOT (CIN * 9)            // 4608
#define CHUNK_CI 64
#define NCHUNK (CIN / CHUNK_CI)   // 8
#define TSTEPS (CHUNK_CI / 4)     // 16 wmma steps per khkw group
#define SG_TOTAL (NCHUNK * 9 * TSTEPS)  // 1152 total wmma steps
#define NTILE 13                  // ceil(196/16)
#define CF_ELEMS (2 * 256 * PR * PR)       // 100352
#define APACK_ELEMS (16 * SG_TOTAL * 64)   // 1179648

// ---------- Kernel 1: bilinear resize features [2,256,96,128] -> cf [2,256,14,14] ----------
__global__ void resize_cf_kernel(const float* __restrict__ feat, float* __restrict__ cf) {
  int idx = blockIdx.x * blockDim.x + threadIdx.x;
  if (idx >= CF_ELEMS) return;
  int ox = idx % PR;
  int t = idx / PR;
  int oy = t % PR;
  int bc = t / PR;  // [0, 512)
  const float* f = feat + (size_t)bc * 96 * 128;
  float sy = fmaxf((oy + 0.5f) * (96.0f / PR) - 0.5f, 0.0f);
  int y0 = min((int)floorf(sy), 95);
  int y1 = min(y0 + 1, 95);
  float wy = sy - (float)y0;
  float sx = fmaxf((ox + 0.5f) * (128.0f / PR) - 0.5f, 0.0f);
  int x0 = min((int)floorf(sx), 127);
  int x1 = min(x0 + 1, 127);
  float wx = sx - (float)x0;
  float r0 = f[y0 * 128 + x0] * (1.0f - wx) + f[y0 * 128 + x1] * wx;
  float r1 = f[y1 * 128 + x0] * (1.0f - wx) + f[y1 * 128 + x1] * wx;
  cf[idx] = r0 * (1.0f - wy) + r1 * wy;
}

// ---------- Kernel 2: pack conv_w [256,512,3,3] into WMMA A-fragment order ----------
// K enumeration: chunk c (64 ci) -> khkw (0..8) -> step t (0..15); step covers ci = 4t..4t+3.
// Pair-interleaved so one global_load_b64 fetches both A VGPRs of a lane:
// apack[((tile*1152 + sg)*32 + lane)*2 + v]
//   = W[tile*16 + (lane&15)][(c*64 + 4t + v + 2*(lane>>4))*9 + khkw],  sg = c*144 + khkw*16 + t
__global__ void pack_w_kernel(const float* __restrict__ w, float* __restrict__ apack) {
  int idx = blockIdx.x * blockDim.x + threadIdx.x;
  if (idx >= APACK_ELEMS) return;
  int v = idx & 1;
  int lane = (idx >> 1) & 31;
  int sg = (idx >> 6) % SG_TOTAL;
  int tile = idx / (SG_TOTAL * 64);
  int c = sg / 144;
  int rem = sg - c * 144;
  int khkw = rem >> 4;
  int t = rem & 15;
  int ci = c * CHUNK_CI + 4 * t + v + 2 * (lane >> 4);
  int k = ci * 9 + khkw;
  int co = tile * 16 + (lane & 15);
  apack[idx] = w[(size_t)co * KTOT + k];
}

// ---------- Kernel 3: implicit-GEMM conv3x3 + bias + BN + ReLU via V_WMMA_F32_16X16X4_F32 ----------
__launch_bounds__(256)
__global__ void conv_wmma_kernel(const float* __restrict__ mf,     // [512,256,14,14]
                                 const float* __restrict__ pb,     // [2,256,4]
                                 const float* __restrict__ cf,     // [2,256,14,14]
                                 const float* __restrict__ apack,  // packed weights
                                 const float* __restrict__ bias,
                                 const float* __restrict__ gamma,
                                 const float* __restrict__ beta,
                                 const float* __restrict__ mean,
                                 const float* __restrict__ var,
                                 float* __restrict__ out) {        // [512,256,14,14]
  __shared__ float xs[CHUNK_CI * 256];  // 64 ci * (16x16 zero-padded tile) = 64 KB
  const int n = blockIdx.y;             // image / box index [0,512)
  const int co0 = blockIdx.x * 128;     // output-channel half
  const int tid = threadIdx.x;
  const int lane = tid & 31;
  const int wave = tid >> 5;
  const int hih = lane >> 4;
  const int nl = lane & 15;

  // ---- per-box crop parameters (uniform; compiler scalarizes) ----
  const int b = n >> 8;
  const float* box = pb + (size_t)n * 4;
  int xa = (int)(box[0] * 0.125f), ya = (int)(box[1] * 0.125f);
  int xb = (int)(box[2] * 0.125f), yb = (int)(box[3] * 0.125f);
  if (xb - xa < 1) { if (xb == PR) xa -= 1; else xb += 1; }
  if (yb - ya < 1) { if (yb == PR) ya -= 1; else yb += 1; }
  const float wn = (float)(xb - xa), hn = (float)(yb - ya);
  const int wlim = xb - xa - 1, hlim = yb - ya - 1;
  const float* cfb = cf + (size_t)b * 256 * PR * PR;

  // ---- per-lane B base offsets (floats): spatial pos + K-row pair fold, constant all kernel ----
  const int krow0 = hih * 2;  // B VGPR0 holds K rows {0,2}, VGPR1 {1,3}
  int yxbk[NTILE];
#pragma unroll
  for (int j = 0; j < NTILE; ++j) {
    int p = 16 * j + nl;
    if (p > 195) p = 195;  // clamp tail tile; garbage cols never stored
    yxbk[j] = (p / 14) * 16 + (p % 14) + krow0 * 256;
  }

  v8f acc[NTILE];
#pragma unroll
  for (int j = 0; j < NTILE; ++j)
#pragma unroll
    for (int r = 0; r < 8; ++r) acc[j][r] = 0.0f;

  const float* apt = apack + (size_t)(blockIdx.x * 8 + wave) * (size_t)SG_TOTAL * 64;
  const float* mfn = mf + (size_t)n * CMID * PR * PR;

  for (int c = 0; c < NCHUNK; ++c) {
    __syncthreads();
    const int cbase = c * CHUNK_CI;
    // ---- stage 64 input channels into zero-padded 16x16 LDS tiles ----
    for (int i = tid; i < CHUNK_CI * 256; i += 256) {
      int cil = i >> 8;
      int slot = i & 255;
      int y = (slot >> 4) - 1;
      int x = (slot & 15) - 1;
      float v = 0.0f;
      if ((unsigned)y < 14u && (unsigned)x < 14u) {
        int cig = cbase + cil;
        if (cig < CMID) {
          v = mfn[((size_t)cig * PR + y) * PR + x];
        } else {
          // on-the-fly crop_resize from cf
          int cc = cig - CMID;
          float sy = fmaxf(((float)y + 0.5f) * hn * (1.0f / PR) - 0.5f, 0.0f);
          int i0 = (int)sy; if (i0 > hlim) i0 = hlim;
          int i1 = i0 + 1;  if (i1 > hlim) i1 = hlim;
          float wy = sy - (float)i0;
          int Y0 = ya + i0, Y1 = ya + i1;
          float sx = fmaxf(((float)x + 0.5f) * wn * (1.0f / PR) - 0.5f, 0.0f);
          int j0 = (int)sx; if (j0 > wlim) j0 = wlim;
          int j1 = j0 + 1;  if (j1 > wlim) j1 = wlim;
          float wx = sx - (float)j0;
          int X0 = xa + j0, X1 = xa + j1;
          const float* fm = cfb + (size_t)cc * PR * PR;
          float v00 = fm[Y0 * PR + X0], v01 = fm[Y0 * PR + X1];
          float v10 = fm[Y1 * PR + X0], v11 = fm[Y1 * PR + X1];
          float r0 = v00 * (1.0f - wy) + v10 * wy;
          float r1 = v01 * (1.0f - wy) + v11 * wy;
          v = r0 * (1.0f - wx) + r1 * wx;
        }
      }
      xs[i] = v;
    }
    // prefetch next chunk's mask_features slab toward the WGP (global_prefetch_b8)
    if (c + 1 < CMID / CHUNK_CI) {
      const float* nx = mfn + (size_t)(cbase + CHUNK_CI) * PR * PR;
      __builtin_prefetch(nx + (size_t)tid * 49, 0, 1);
      __builtin_prefetch(nx + (size_t)tid * 49 + 32, 0, 1);
    }
    __syncthreads();

    // ---- GEMM over this K chunk: 9 (kh,kw) groups x 16 steps of 16x16x4 ----
    const float* apc = apt + (size_t)(c * 144) * 64;
    for (int khkw = 0; khkw < 9; ++khkw) {
      const int kh = khkw / 3;
      const int khoff = kh * 16 + (khkw - kh * 3);  // kh*16 + kw
      int base[NTILE];
#pragma unroll
      for (int j = 0; j < NTILE; ++j) base[j] = yxbk[j] + khoff;
      const float* apk = apc + (size_t)(khkw * TSTEPS) * 64;
#pragma unroll
      for (int t = 0; t < TSTEPS; ++t) {
        v2f a = *(const v2f*)(apk + t * 64 + lane * 2);  // one global_load_b64 per step
#pragma unroll
        for (int j = 0; j < NTILE; ++j) {
          v2f bb;
          bb.x = xs[base[j] + t * 1024];        // ds_load_2addr_stride64_b32 pair
          bb.y = xs[base[j] + t * 1024 + 256];
          acc[j] = __builtin_amdgcn_wmma_f32_16x16x4_f32(false, a, false, bb, (short)0,
                                                         acc[j], false, false);
        }
      }
    }
  }

  // ---- fused bias + BN + ReLU epilogue; C/D layout: VGPR r -> row r + 8*(lane>=16) ----
  float mulv[8], addv[8];
#pragma unroll
  for (int r = 0; r < 8; ++r) {
    int co = co0 + wave * 16 + hih * 8 + r;
    float sc = gamma[co] * rsqrtf(var[co] + 1e-5f);
    mulv[r] = sc;
    addv[r] = (bias[co] - mean[co]) * sc + beta[co];
  }
  float* outn = out + ((size_t)n * 256 + co0 + wave * 16 + hih * 8) * 196;
#pragma unroll
  for (int j = 0; j < NTILE; ++j) {
    int p = 16 * j + nl;
    if (p < 196) {
#pragma unroll
      for (int r = 0; r < 8; ++r) {
        float vv = acc[j][r] * mulv[r] + addv[r];
        outn[(size_t)r * 196 + p] = fmaxf(vv, 0.0f);
      }
    }
  }
}

extern "C" void kernel_launch(void* const* d_in, const int* in_sizes, int n_in,
                              void* d_out, int out_size, void* d_ws, size_t ws_size,
                              hipStream_t stream) {
  const float* feat = (const float*)d_in[0];
  const float* pb   = (const float*)d_in[1];
  const float* mf   = (const float*)d_in[2];
  const float* cw   = (const float*)d_in[3];
  const float* cb   = (const float*)d_in[4];
  const float* gm   = (const float*)d_in[5];
  const float* bt   = (const float*)d_in[6];
  const float* mn   = (const float*)d_in[7];
  const float* vr   = (const float*)d_in[8];
  float* out = (float*)d_out;

  float* cf = (float*)d_ws;            // 100352 floats (401 KB)
  float* apack = cf + CF_ELEMS;        // 1179648 floats (4.7 MB)

  hipLaunchKernelGGL(resize_cf_kernel, dim3((CF_ELEMS + 255) / 256), dim3(256), 0, stream,
                     feat, cf);
  hipLaunchKernelGGL(pack_w_kernel, dim3((APACK_ELEMS + 255) / 256), dim3(256), 0, stream,
                     cw, apack);
  hipLaunchKernelGGL(conv_wmma_kernel, dim3(2, 512), dim3(256), 0, stream,
                     mf, pb, cf, apack, cb, gm, bt, mn, vr, out);
}